// RegionDiceLoss_2800318677062
// MI455X (gfx1250) — compile-verified
//
#include <hip/hip_runtime.h>

// Problem geometry (fixed by the reference)
#define NUM_REGIONS 32
#define NBINS       33              // regions 0..32 (0 = ignore at the end)
#define BVOL        4
#define VOLN        (160*192*192)   // 5,898,240 elements per volume
#define WPB         128             // workgroups per batch volume
#define COLS        80              // padded partial cols: [0..32]=sum, [33..65]=cnt, rest 0
#define THREADS     256
#define NWAVES      (THREADS/32)
#define BINSTRIDE   72              // per-wave bin stride (padded past 66)
#define EPS         1e-8f

typedef float v2f __attribute__((ext_vector_type(2)));
typedef float v8f __attribute__((ext_vector_type(8)));

// ---------------------------------------------------------------------------
// Pass 1: streaming sigmoid + segmented accumulation into per-wave LDS bins.
// Each workgroup owns a contiguous 46080-element slice of one batch volume.
// Writes an 80-column fp32 partial row to ws[wgid][0..79].
// ---------------------------------------------------------------------------
__global__ __launch_bounds__(THREADS) void region_partials_kernel(
    const float* __restrict__ x, const int* __restrict__ seg,
    float* __restrict__ part)
{
    __shared__ float bins[NWAVES * BINSTRIDE];
    const int tid = threadIdx.x;
    for (int i = tid; i < NWAVES * BINSTRIDE; i += THREADS) bins[i] = 0.f;
    __syncthreads();

    const int wgid = blockIdx.x;            // 0 .. BVOL*WPB-1
    const int b    = wgid / WPB;
    const int wgi  = wgid % WPB;
    const long long base = (long long)b * VOLN + (long long)wgi * (VOLN / WPB);
    const int nvec = (VOLN / WPB) / 4;      // float4 / int4 chunks

    const float4* __restrict__ x4 = (const float4*)(x + base);
    const int4*   __restrict__ s4 = (const int4*)(seg + base);
    float* wbins = &bins[(tid >> 5) * BINSTRIDE];   // wave32: per-wave private bins

    for (int j = tid; j < nvec; j += THREADS) {
        float4 xv = x4[j];                  // global_load_b128
        int4   rv = s4[j];                  // global_load_b128
        float p0 = 1.f / (1.f + __expf(-xv.x));
        float p1 = 1.f / (1.f + __expf(-xv.y));
        float p2 = 1.f / (1.f + __expf(-xv.z));
        float p3 = 1.f / (1.f + __expf(-xv.w));
        if ((unsigned)rv.x <= NUM_REGIONS) { atomicAdd(&wbins[rv.x], p0); atomicAdd(&wbins[NBINS + rv.x], 1.f); }
        if ((unsigned)rv.y <= NUM_REGIONS) { atomicAdd(&wbins[rv.y], p1); atomicAdd(&wbins[NBINS + rv.y], 1.f); }
        if ((unsigned)rv.z <= NUM_REGIONS) { atomicAdd(&wbins[rv.z], p2); atomicAdd(&wbins[NBINS + rv.z], 1.f); }
        if ((unsigned)rv.w <= NUM_REGIONS) { atomicAdd(&wbins[rv.w], p3); atomicAdd(&wbins[NBINS + rv.w], 1.f); }
    }
    __syncthreads();

    // Cross-wave reduce the 8 private copies, emit padded 80-col partial row.
    if (tid < COLS) {
        float s = 0.f;
        if (tid < 2 * NBINS) {
            #pragma unroll
            for (int w = 0; w < NWAVES; ++w) s += bins[w * BINSTRIDE + tid];
        }
        part[(long long)wgid * COLS + tid] = s;
    }
}

// ---------------------------------------------------------------------------
// Pass 2: single wave32. Column-reduce the WPB x COLS partial matrix per batch
// with V_WMMA_F32_16X16X4_F32 (A = all-ones 16x4 => D[m][n] = sum_k B[k][n]),
// then compute the dice scalar on lane 0.
// ---------------------------------------------------------------------------
__global__ __launch_bounds__(32) void region_finalize_kernel(
    const float* __restrict__ part, float* __restrict__ out)
{
    __shared__ float tot[COLS];
    const int lane = threadIdx.x;
    const int n    = lane & 15;     // column within 16-wide tile (lane mod 16)
    const int half = lane >> 4;     // K-half selector for B operand
    float loss_sum = 0.f;

    for (int b = 0; b < BVOL; ++b) {
        const float* pb = part + (long long)b * WPB * COLS;
        for (int t = 0; t < COLS / 16; ++t) {           // 5 column tiles of 16
            v8f c = {};
            for (int kk = 0; kk < WPB / 4; ++kk) {      // K=4 rows per WMMA
                const int wg0 = kk * 4 + 2 * half;      // lanes 16-31 take the other K-half
                v2f bm;
                bm.x = pb[(long long)(wg0 + 0) * COLS + t * 16 + n];
                bm.y = pb[(long long)(wg0 + 1) * COLS + t * 16 + n];
                v2f am; am.x = 1.f; am.y = 1.f;         // all-ones A: pure column sum
                c = __builtin_amdgcn_wmma_f32_16x16x4_f32(
                        /*neg_a=*/false, am, /*neg_b=*/false, bm,
                        /*c_mod=*/(short)0, c, /*reuse_a=*/false, /*reuse_b=*/false);
            }
            // D row M=0 lives in c[0] on lanes 0..15, N = lane.
            if (lane < 16) tot[t * 16 + lane] = c[0];
        }
        __syncthreads();
        if (lane == 0) {
            float acc = 0.f; int nv = 0;
            for (int r = 1; r <= NUM_REGIONS; ++r) {
                float xs = tot[r];
                float cn = tot[NBINS + r];
                if (cn > 0.f) { acc += 2.f * xs / (xs + cn + EPS); ++nv; }
            }
            loss_sum += (nv > 0) ? (1.f - acc / (float)nv) : 1.f;
        }
        __syncthreads();
    }
    if (lane == 0) out[0] = loss_sum * (1.f / (float)BVOL);
}

// ---------------------------------------------------------------------------
extern "C" void kernel_launch(void* const* d_in, const int* in_sizes, int n_in,
                              void* d_out, int out_size, void* d_ws, size_t ws_size,
                              hipStream_t stream) {
    (void)in_sizes; (void)n_in; (void)out_size; (void)ws_size;
    const float* x   = (const float*)d_in[0];
    const int*   seg = (const int*)d_in[1];
    float*       out = (float*)d_out;
    float*       part = (float*)d_ws;   // BVOL*WPB*COLS floats = 160 KiB

    region_partials_kernel<<<BVOL * WPB, THREADS, 0, stream>>>(x, seg, part);
    region_finalize_kernel<<<1, 32, 0, stream>>>(part, out);
}